// HIndexer_44135083933925
// MI455X (gfx1250) — compile-verified
//
#include <hip/hip_runtime.h>
#include <hip/hip_bf16.h>

typedef __attribute__((ext_vector_type(2))) float v2f;
typedef __attribute__((ext_vector_type(8))) float v8f;

#define D_IN_MAX 512
#define D_ENC_MAX 256

// ---------------------------------------------------------------------------
// Kernel A: q = u @ Wu  (d_enc) ; v = Wi @ q  (d_in).  One block, tiny work.
// ---------------------------------------------------------------------------
__global__ void encode_kernel(const float* __restrict__ u,
                              const float* __restrict__ Wu,
                              const float* __restrict__ Wi,
                              float* __restrict__ v_out,
                              int d_in, int d_enc) {
    __shared__ float su[D_IN_MAX];
    __shared__ float sq[D_ENC_MAX];
    const int tid = threadIdx.x;          // 256 threads
    const int nt  = blockDim.x;

    for (int i = tid; i < d_in; i += nt) su[i] = u[i];
    __syncthreads();

    // q[j] = sum_i u[i] * Wu[i*d_enc + j]   (coalesced across j)
    for (int j = tid; j < d_enc; j += nt) {
        float acc = 0.0f;
        for (int i = 0; i < d_in; ++i) acc += su[i] * Wu[(long)i * d_enc + j];
        sq[j] = acc;
    }
    __syncthreads();

    // v[i] = sum_j Wi[i*d_enc + j] * q[j]
    for (int i = tid; i < d_in; i += nt) {
        const float* wrow = Wi + (long)i * d_enc;
        float acc = 0.0f;
        for (int j = 0; j < d_enc; ++j) acc += wrow[j] * sq[j];
        v_out[i] = acc;
    }
}

// ---------------------------------------------------------------------------
// Kernel B: sims[r] = dot(items[r, :], v)  via V_WMMA_F32_16X16X4_F32.
// One wave per 16-row tile; B-operand carries v broadcast to all 16 columns,
// so D[m][n] == sims[rowBase+m] for every n; extract column 0 (lanes 0 & 16).
// ISA layouts (wave32):
//   A (16x4 f32, 2 VGPR): lane l -> row (l&15), K = k0 + (l>>4)*2 + {0,1}
//   B (4x16 f32, 2 VGPR): lane l -> col (l&15), K = k0 + (l>>4)*2 + {0,1}
//   D (16x16 f32, 8 VGPR): vgpr g, lane l<16 -> (M=g,   N=l)
//                          vgpr g, lane l>=16 -> (M=g+8, N=l-16)
// ---------------------------------------------------------------------------
__global__ void sims_wmma_kernel(const float* __restrict__ items,
                                 const float* __restrict__ v,
                                 float* __restrict__ sims,
                                 int n, int d_in) {
    __shared__ float vs[D_IN_MAX];
    const int tid = threadIdx.x;
    for (int i = tid; i < d_in; i += blockDim.x) vs[i] = v[i];
    __syncthreads();

    const int lane       = tid & 31;
    const int waveInBlk  = tid >> 5;
    const long waveGlob  = (long)blockIdx.x * (blockDim.x >> 5) + waveInBlk;
    const long rowBase   = waveGlob * 16;
    if (rowBase >= n) return;                 // wave-uniform exit: EXEC stays full

    const int  m    = lane & 15;
    const int  koff = (lane >> 4) * 2;        // lanes 16-31 hold the K+2 pair
    long row = rowBase + m;
    if (row >= n) row = n - 1;                // safe clamp for a ragged tail
    const float* arow = items + row * (long)d_in + koff;
    const float* brow = vs + koff;

    v8f c = {};
    #pragma unroll 4
    for (int k0 = 0; k0 < d_in; k0 += 4) {
        v2f a = *(const v2f*)(arow + k0);     // global_load_b64, streaming
        v2f b = *(const v2f*)(brow + k0);     // LDS broadcast, conflict-free
        // 8 args: (neg_a, A, neg_b, B, c_mod, C, reuse_a, reuse_b)
        c = __builtin_amdgcn_wmma_f32_16x16x4_f32(
                false, a, false, b, (short)0, c, false, false);
    }

    // Column 0 of D: lane 0 has M=0..7 in c[0..7], lane 16 has M=8..15.
    if (lane == 0 || lane == 16) {
        float* outp = sims + rowBase + (lane >> 4) * 8;
        const long lim = n - (rowBase + (lane >> 4) * 8);
        #pragma unroll
        for (int g = 0; g < 8; ++g)
            if (g < lim) outp[g] = c[g];
    }
}

// ---------------------------------------------------------------------------
// Kernel C: gather lamb sampled sims, rank-select the threshold element
// (== jnp.sort(sampled)[min(k*lamb//n, lamb-1)]) by O(lamb^2) counting.
// ---------------------------------------------------------------------------
__global__ void threshold_kernel(const float* __restrict__ sims,
                                 const int* __restrict__ sample_idx,
                                 const int* __restrict__ k_ptr,
                                 int lamb, int n,
                                 float* __restrict__ t_out) {
    extern __shared__ float s[];
    const int tid = threadIdx.x;
    const int nt  = blockDim.x;

    for (int i = tid; i < lamb; i += nt) s[i] = sims[sample_idx[i]];
    __syncthreads();

    const long long k  = (long long)k_ptr[0];
    long long ti = k * (long long)lamb / (long long)n;
    if (ti > lamb - 1) ti = lamb - 1;

    for (int i = tid; i < lamb; i += nt) {
        const float xi = s[i];
        int rank = 0;
        for (int j = 0; j < lamb; ++j) {
            const float xj = s[j];
            rank += (xj < xi) || (xj == xi && j < i);   // stable rank
        }
        if (rank == (int)ti) *t_out = xi;               // exactly one writer
    }
}

// ---------------------------------------------------------------------------
// Kernel D: in-place masked copy: out[i] = sims[i] > t ? sims[i] : 0
// ---------------------------------------------------------------------------
__global__ void select_kernel(float* __restrict__ out,
                              const float* __restrict__ t_ptr, int n) {
    const int i = blockIdx.x * blockDim.x + threadIdx.x;
    if (i < n) {
        const float t = *t_ptr;
        const float s = out[i];
        out[i] = (s > t) ? s : 0.0f;
    }
}

// ---------------------------------------------------------------------------
extern "C" void kernel_launch(void* const* d_in, const int* in_sizes, int n_in,
                              void* d_out, int out_size, void* d_ws, size_t ws_size,
                              hipStream_t stream) {
    const float* user  = (const float*)d_in[0];   // [d_in]
    const float* items = (const float*)d_in[1];   // [n, d_in]
    const float* Wu    = (const float*)d_in[2];   // [d_in, d_enc]
    const float* Wi    = (const float*)d_in[3];   // [d_in, d_enc]
    const int*   sidx  = (const int*)d_in[4];     // [lamb]
    const int*   k_p   = (const int*)d_in[6];     // scalar

    const int d_in_dim = in_sizes[0];             // 512
    const int n        = in_sizes[1] / d_in_dim;  // 200000
    const int d_enc    = in_sizes[2] / d_in_dim;  // 256
    const int lamb     = in_sizes[4];             // 1000

    float* out = (float*)d_out;                   // doubles as sims buffer
    float* v   = (float*)d_ws;                    // [d_in] folded weight vector
    float* t   = v + d_in_dim;                    // [1] threshold

    // 1) fold both projections into one 512-vector
    encode_kernel<<<1, 256, 0, stream>>>(user, Wu, Wi, v, d_in_dim, d_enc);

    // 2) bandwidth-bound GEMV via WMMA: one wave per 16 rows
    const int waves  = (n + 15) / 16;
    const int wPerB  = 8;                         // 256 threads/block
    const int blocks = (waves + wPerB - 1) / wPerB;
    sims_wmma_kernel<<<blocks, 256, 0, stream>>>(items, v, out, n, d_in_dim);

    // 3) sampled rank-select threshold (reads device scalar k)
    threshold_kernel<<<1, 1024, (size_t)lamb * sizeof(float), stream>>>(
        out, sidx, k_p, lamb, n, t);

    // 4) masked copy, in place
    select_kernel<<<(n + 255) / 256, 256, 0, stream>>>(out, t, n);
}